// S2SAttentionDecoder_944892805532
// MI455X (gfx1250) — compile-verified
//
#include <hip/hip_runtime.h>
#include <hip/hip_bf16.h>

// ---------------------------------------------------------------------------
// S2S attention decoder for MI455X (gfx1250).
// All GEMMs run through V_WMMA_F32_16X16X4_F32 (full fp32 accuracy; the
// workload is HBM/L2 bound at ~0.5 GB of traffic, so low-precision WMMA
// would not help and could break validation).
// Round 2: 2x2 register-blocked WMMA tiles (32x32 per wave) -> 1 global load
// per WMMA instead of 2, halving fragment bandwidth on the bandwidth-bound
// logits GEMM, with 4 independent accumulator chains per wave.
// ---------------------------------------------------------------------------

#define B_  64
#define TD_ 32
#define TE_ 64
#define H_  1024
#define E_  512
#define V_  32000

typedef __attribute__((ext_vector_type(2))) float v2f;
typedef __attribute__((ext_vector_type(8))) float v8f;

// ---------------------------------------------------------------------------
// Generic WMMA GEMM:  C[M,N] = A[M,K] * W[N,K]^T (+ bias[N] if non-null)
// A row-major (lda=K), W row-major (ldw=K), C row-major (ldc=N).
// Block = 128 threads = 4 waves. Each wave computes a 32(M) x 32(N) block as
// 2x2 WMMA 16x16 tiles; block covers 32(M) x 128(N).
// Requirements (all call sites satisfy): M%32==0, N%128==0, K%32==0.
//
// fp32 WMMA operand layout (ISA 7.12.2):
//   A 16x4 : lanes 0-15 rows 0-15 k={0,1}; lanes 16-31 rows 0-15 k={2,3}
//   B 4x16 : lanes 0-15 cols 0-15 k={0,1}; lanes 16-31 cols 0-15 k={2,3}
//   C 16x16: VGPR v -> row v (lanes 0-15) / row v+8 (lanes 16-31), col lane&15
// ---------------------------------------------------------------------------
__global__ __launch_bounds__(128)
void wmma_gemm_f32_kernel(const float* __restrict__ A,
                          const float* __restrict__ W,
                          const float* __restrict__ bias,
                          float* __restrict__ C,
                          int M, int N, int K) {
  const int lane = threadIdx.x & 31;
  const int wave = threadIdx.x >> 5;
  const int n0   = (blockIdx.x * 4 + wave) << 5;   // 32-wide N block per wave
  const int m0   = blockIdx.y << 5;                // 32-tall M block
  const int l15  = lane & 15;
  const int koff = (lane >> 4) << 1;               // 0 for lanes 0-15, 2 for 16-31

  const float* a0 = A + (size_t)(m0 + l15) * K + koff;
  const float* a1 = a0 + (size_t)16 * K;
  const float* w0 = W + (size_t)(n0 + l15) * K + koff;
  const float* w1 = w0 + (size_t)16 * K;

  v8f acc00 = {0.f,0.f,0.f,0.f,0.f,0.f,0.f,0.f};
  v8f acc01 = acc00, acc10 = acc00, acc11 = acc00;

  for (int k = 0; k < K; k += 32) {
    // Prefetch 2 cachelines ahead on each of the 4 streams
    // (emits global_prefetch_b8; speculative, OOB-safe).
    __builtin_prefetch(a0 + k + 64, 0, 1);
    __builtin_prefetch(a1 + k + 64, 0, 1);
    __builtin_prefetch(w0 + k + 64, 0, 1);
    __builtin_prefetch(w1 + k + 64, 0, 1);
#pragma unroll
    for (int kk = 0; kk < 32; kk += 4) {
      v2f av0 = *(const v2f*)(a0 + k + kk);
      v2f av1 = *(const v2f*)(a1 + k + kk);
      v2f bv0 = *(const v2f*)(w0 + k + kk);
      v2f bv1 = *(const v2f*)(w1 + k + kk);
      acc00 = __builtin_amdgcn_wmma_f32_16x16x4_f32(false, av0, false, bv0,
                                                    (short)0, acc00, false, false);
      acc01 = __builtin_amdgcn_wmma_f32_16x16x4_f32(false, av0, false, bv1,
                                                    (short)0, acc01, false, false);
      acc10 = __builtin_amdgcn_wmma_f32_16x16x4_f32(false, av1, false, bv0,
                                                    (short)0, acc10, false, false);
      acc11 = __builtin_amdgcn_wmma_f32_16x16x4_f32(false, av1, false, bv1,
                                                    (short)0, acc11, false, false);
    }
  }

  const int   cn0 = n0 + l15;
  const int   cn1 = cn0 + 16;
  const int   cm0 = m0 + ((lane >> 4) << 3);
  const int   cm1 = cm0 + 16;
  const float bv0 = bias ? bias[cn0] : 0.0f;
  const float bv1 = bias ? bias[cn1] : 0.0f;
#pragma unroll
  for (int v = 0; v < 8; ++v) {
    C[(size_t)(cm0 + v) * N + cn0] = acc00[v] + bv0;
    C[(size_t)(cm0 + v) * N + cn1] = acc01[v] + bv1;
    C[(size_t)(cm1 + v) * N + cn0] = acc10[v] + bv0;
    C[(size_t)(cm1 + v) * N + cn1] = acc11[v] + bv1;
  }
}

// ---------------------------------------------------------------------------
// Additive-attention scores: score[b,te] = sum_h tanh(q[b,h]+keys[b,te,h])*wv[h]
// with length masking.  One 256-thread block per (b, te).
// ---------------------------------------------------------------------------
__global__ __launch_bounds__(256)
void attn_scores_kernel(const float* __restrict__ q,
                        const float* __restrict__ keys,
                        const float* __restrict__ wv,
                        const int* __restrict__ vlen,
                        float* __restrict__ scores) {
  const int b  = blockIdx.x;
  const int te = blockIdx.y;
  __shared__ float red[256];

  const float* qb = q + (size_t)b * H_;
  const float* kb = keys + ((size_t)b * TE_ + te) * H_;
  float s = 0.f;
  for (int h = threadIdx.x; h < H_; h += 256)
    s += tanhf(qb[h] + kb[h]) * wv[h];
  red[threadIdx.x] = s;
  __syncthreads();
  for (int off = 128; off > 0; off >>= 1) {
    if (threadIdx.x < off) red[threadIdx.x] += red[threadIdx.x + off];
    __syncthreads();
  }
  if (threadIdx.x == 0)
    scores[b * TE_ + te] = (te < vlen[b]) ? red[0] : -1e6f;
}

// ---------------------------------------------------------------------------
// Softmax over TE, context vector, and GRU input assembly:
//   inp[b, 0:H)   = sum_te softmax(scores[b])[te] * enc[b,te,:]
//   inp[b, H:H+E) = emb[X[b,t]]
// One 256-thread block per batch element b.
// ---------------------------------------------------------------------------
__global__ __launch_bounds__(256)
void softmax_ctx_embed_kernel(const float* __restrict__ scores,
                              const float* __restrict__ enc,
                              const int* __restrict__ X,
                              const float* __restrict__ emb,
                              float* __restrict__ inp,
                              int t) {
  const int b = blockIdx.x;
  __shared__ float w[TE_];
  __shared__ float sc[TE_];

  if (threadIdx.x < TE_) sc[threadIdx.x] = scores[b * TE_ + threadIdx.x];
  __syncthreads();
  if (threadIdx.x == 0) {
    float mx = -1e30f;
    for (int i = 0; i < TE_; ++i) mx = fmaxf(mx, sc[i]);
    float sum = 0.f;
    for (int i = 0; i < TE_; ++i) { float e = expf(sc[i] - mx); w[i] = e; sum += e; }
    float inv = 1.f / sum;
    for (int i = 0; i < TE_; ++i) w[i] *= inv;
  }
  __syncthreads();

  const float* encb = enc + (size_t)b * TE_ * H_;
  float* inpb = inp + (size_t)b * (H_ + E_);
  for (int h = threadIdx.x; h < H_; h += 256) {
    float c = 0.f;
#pragma unroll 8
    for (int te = 0; te < TE_; ++te) c += w[te] * encb[(size_t)te * H_ + h];
    inpb[h] = c;
  }

  const int tok = X[b * TD_ + t];
  const float* eb = emb + (size_t)tok * E_;
  for (int e = threadIdx.x; e < E_; e += 256) inpb[H_ + e] = eb[e];
}

// ---------------------------------------------------------------------------
// GRU gate elementwise (PyTorch order r,z,n):
//   r = sig(ir+hr); z = sig(iz+hz); n = tanh(in + r*hn)
//   h_new = (1-z)*n + z*h
// Optionally records h_new into outs[b,t,:] (top layer only).
// ---------------------------------------------------------------------------
__global__ __launch_bounds__(256)
void gru_elem_kernel(const float* __restrict__ gi,
                     const float* __restrict__ gh,
                     float* __restrict__ h,
                     float* __restrict__ outs,
                     int t) {
  const int i  = blockIdx.x * 256 + threadIdx.x;   // 0 .. B*H-1
  const int b  = i >> 10;                          // H == 1024
  const int hh = i & (H_ - 1);
  const float* gib = gi + (size_t)b * 3 * H_;
  const float* ghb = gh + (size_t)b * 3 * H_;

  const float ir = gib[hh],        hr = ghb[hh];
  const float iz = gib[H_ + hh],   hz = ghb[H_ + hh];
  const float in = gib[2*H_ + hh], hn = ghb[2*H_ + hh];

  const float r = 1.f / (1.f + expf(-(ir + hr)));
  const float z = 1.f / (1.f + expf(-(iz + hz)));
  const float n = tanhf(in + r * hn);
  const float hnew = (1.f - z) * n + z * h[i];

  h[i] = hnew;
  if (outs) outs[((size_t)b * TD_ + t) * H_ + hh] = hnew;
}

__global__ __launch_bounds__(256)
void copy_f32_kernel(const float* __restrict__ src, float* __restrict__ dst, int n) {
  const int i = blockIdx.x * 256 + threadIdx.x;
  if (i < n) dst[i] = src[i];
}

// ---------------------------------------------------------------------------
extern "C" void kernel_launch(void* const* d_in, const int* in_sizes, int n_in,
                              void* d_out, int out_size, void* d_ws, size_t ws_size,
                              hipStream_t stream) {
  const int*   X    = (const int*)  d_in[0];
  const float* enc  = (const float*)d_in[1];
  const float* hid0 = (const float*)d_in[2];
  const int*   vlen = (const int*)  d_in[3];
  const float* emb  = (const float*)d_in[4];
  const float* Wq   = (const float*)d_in[5];
  const float* Wk   = (const float*)d_in[6];
  const float* wv   = (const float*)d_in[7];
  const float* Wih0 = (const float*)d_in[8];
  const float* Whh0 = (const float*)d_in[9];
  const float* bih0 = (const float*)d_in[10];
  const float* bhh0 = (const float*)d_in[11];
  const float* Wih1 = (const float*)d_in[12];
  const float* Whh1 = (const float*)d_in[13];
  const float* bih1 = (const float*)d_in[14];
  const float* bhh1 = (const float*)d_in[15];
  const float* linW = (const float*)d_in[16];
  const float* linb = (const float*)d_in[17];

  // Workspace carve-up (~29.5 MB of fp32). h0/h1 kept adjacent so hidden state
  // copies match the [L,B,H] layout with a single memcpy-style kernel.
  float* ws   = (float*)d_ws;
  float* keys = ws;  ws += (size_t)B_ * TE_ * H_;       // 4096x1024
  float* q    = ws;  ws += (size_t)B_ * H_;
  float* scr  = ws;  ws += (size_t)B_ * TE_;
  float* inp  = ws;  ws += (size_t)B_ * (H_ + E_);
  float* gi0  = ws;  ws += (size_t)B_ * 3 * H_;
  float* gh0  = ws;  ws += (size_t)B_ * 3 * H_;
  float* gi1  = ws;  ws += (size_t)B_ * 3 * H_;
  float* gh1  = ws;  ws += (size_t)B_ * 3 * H_;
  float* h0   = ws;  ws += (size_t)B_ * H_;
  float* h1   = ws;  ws += (size_t)B_ * H_;             // contiguous after h0
  float* outs = ws;  ws += (size_t)B_ * TD_ * H_;

  float* logits = (float*)d_out;
  float* hidf   = logits + (size_t)B_ * TD_ * V_;

  // h0|h1 <- hidden_state[0]|hidden_state[1]
  copy_f32_kernel<<<(2 * B_ * H_ + 255) / 256, 256, 0, stream>>>(hid0, h0, 2 * B_ * H_);

  // keys = enc_outputs @ Wk^T  (M=4096, N=1024, K=1024) — done once.
  wmma_gemm_f32_kernel<<<dim3(H_ / 128, (B_ * TE_) / 32), 128, 0, stream>>>(
      enc, Wk, nullptr, keys, B_ * TE_, H_, H_);

  for (int t = 0; t < TD_; ++t) {
    // q = h1 @ Wq^T   (M=64, N=1024, K=1024)
    wmma_gemm_f32_kernel<<<dim3(H_ / 128, B_ / 32), 128, 0, stream>>>(
        h1, Wq, nullptr, q, B_, H_, H_);

    attn_scores_kernel<<<dim3(B_, TE_), 256, 0, stream>>>(q, keys, wv, vlen, scr);
    softmax_ctx_embed_kernel<<<B_, 256, 0, stream>>>(scr, enc, X, emb, inp, t);

    // Layer 0: gi0 = [ctx|x] @ Wih0^T + bih0 (K=1536); gh0 = h0 @ Whh0^T + bhh0
    wmma_gemm_f32_kernel<<<dim3(3 * H_ / 128, B_ / 32), 128, 0, stream>>>(
        inp, Wih0, bih0, gi0, B_, 3 * H_, H_ + E_);
    wmma_gemm_f32_kernel<<<dim3(3 * H_ / 128, B_ / 32), 128, 0, stream>>>(
        h0, Whh0, bhh0, gh0, B_, 3 * H_, H_);
    gru_elem_kernel<<<B_ * H_ / 256, 256, 0, stream>>>(gi0, gh0, h0, nullptr, t);

    // Layer 1: gi1 = h0 @ Wih1^T + bih1; gh1 = h1 @ Whh1^T + bhh1
    wmma_gemm_f32_kernel<<<dim3(3 * H_ / 128, B_ / 32), 128, 0, stream>>>(
        h0, Wih1, bih1, gi1, B_, 3 * H_, H_);
    wmma_gemm_f32_kernel<<<dim3(3 * H_ / 128, B_ / 32), 128, 0, stream>>>(
        h1, Whh1, bhh1, gh1, B_, 3 * H_, H_);
    gru_elem_kernel<<<B_ * H_ / 256, 256, 0, stream>>>(gi1, gh1, h1, outs, t);
  }

  // logits = outs @ linW^T + linb  (M=2048, N=32000, K=1024) — the big one.
  wmma_gemm_f32_kernel<<<dim3(V_ / 128, (B_ * TD_) / 32), 128, 0, stream>>>(
      outs, linW, linb, logits, B_ * TD_, V_, H_);

  // hid_final = [h0 | h1]
  copy_f32_kernel<<<(2 * B_ * H_ + 255) / 256, 256, 0, stream>>>(h0, hidf, 2 * B_ * H_);
}